// SegmentAwarePool_20220706029799
// MI455X (gfx1250) — compile-verified
//
#include <hip/hip_runtime.h>
#include <hip/hip_bf16.h>

typedef __attribute__((ext_vector_type(2))) float v2f;
typedef __attribute__((ext_vector_type(8))) float v8f;

#define B_ 128
#define S_ 512
#define H_ 1024

// ---------------------------------------------------------------------------
// Phase 1: per-batch segment metadata.
// One thread per batch row scans input_ids / attention_mask serially (512
// elements -> trivial cost vs the 256 MB stream in phase 2).
// meta[b*8 + {0..5}] = {t0, t1, l0, l1, tcnt, lcnt}
// title window = [t0, t1), lead window = [l0, l1); empty windows -> cnt = 0.
// ---------------------------------------------------------------------------
__global__ void seg_meta_kernel(const int* __restrict__ input_ids,
                                const int* __restrict__ attn_mask,
                                const int* __restrict__ sep_tok,
                                int* __restrict__ meta) {
  int b = blockIdx.x * blockDim.x + threadIdx.x;
  if (b >= B_) return;
  const int sep = sep_tok[0];
  const int* ids = input_ids + (size_t)b * S_;
  const int* am  = attn_mask + (size_t)b * S_;

  int num = 0, p1 = 0, p2 = 0, p3 = 0, ml = 0;
  for (int s = 0; s < S_; ++s) {
    int id = ids[s];
    if (id == sep) {
      ++num;
      if (num == 1)      p1 = s;
      else if (num == 2) p2 = s;
      else if (num == 3) p3 = s;
    }
    ml += am[s];
  }
  const bool valid = (num >= 2);
  const int lead_end = (num >= 3) ? p3 : ml;

  int t0 = 1;
  int t1 = valid ? p1 : t0;
  int l0 = p2 + 1;
  int l1 = valid ? lead_end : l0;
  if (t1 < t0) t1 = t0;
  if (l1 < l0) l1 = l0;
  if (t1 > S_) t1 = S_;
  if (l1 > S_) l1 = S_;

  int* m = meta + b * 8;
  m[0] = t0; m[1] = t1;
  m[2] = l0; m[3] = l1;
  m[4] = t1 - t0;   // title count
  m[5] = l1 - l0;   // lead count
}

// ---------------------------------------------------------------------------
// Phase 2: masked sums via V_WMMA_F32_16X16X4_F32.
// One wave32 per (batch, 16-wide H tile). D(16x16) accumulated over S in K=4
// steps: A = 0/1 window weights (row 0 = title, row 1 = lead), B = hidden
// tile. VGPR layouts per CDNA5 ISA Ch7.12.2:
//   A 16x4 : lane L holds M = L&15, VGPR r holds K = (L>=16 ? 2 : 0) + r
//   B 4x16 : lane L holds N = L&15, VGPR r holds K = (L>=16 ? 2 : 0) + r
//   D 16x16: lanes 0-15 hold N = lane; VGPR r holds M = r (rows 0/1 wanted)
//
// The A weights are step functions of s: constant on every interval cut by
// {t0,t1,l0,l1}. The K loop is split into constant-weight runs (weights
// hoisted -> body is just 2 loads + 1 WMMA) plus <=4 boundary-straddling
// K-steps using per-element weights. All interval logic is SGPR-scalar
// (readfirstlane), so EXEC stays all-1s around every WMMA.
// ---------------------------------------------------------------------------
__global__ void __launch_bounds__(32)
seg_pool_wmma_kernel(const float* __restrict__ X,
                     const int* __restrict__ meta,
                     float* __restrict__ out) {
  const int b     = blockIdx.y;
  const int hbase = blockIdx.x * 16;
  const int lane  = threadIdx.x;        // 0..31
  const int n     = lane & 15;          // A row M / B col N / D col N
  const int koff  = (lane >> 4) * 2;    // K offset for this lane half

  const int* m = meta + b * 8;
  const int t0   = __builtin_amdgcn_readfirstlane(m[0]);
  const int t1   = __builtin_amdgcn_readfirstlane(m[1]);
  const int l0   = __builtin_amdgcn_readfirstlane(m[2]);
  const int l1   = __builtin_amdgcn_readfirstlane(m[3]);
  const int tcnt = __builtin_amdgcn_readfirstlane(m[4]);
  const int lcnt = __builtin_amdgcn_readfirstlane(m[5]);

  // Per-lane A-row window: row 0 -> title, row 1 -> lead, rows 2..15 empty.
  const int      lo   = (n == 0) ? t0 : (n == 1) ? l0 : 0;
  const unsigned span = (n == 0) ? (unsigned)(t1 - t0)
                      : (n == 1) ? (unsigned)(l1 - l0) : 0u;

  const float* p = X + ((size_t)b * S_ + koff) * H_ + hbase + n;

  float tsum, lsum;
#if defined(__gfx1250__) && __has_builtin(__builtin_amdgcn_wmma_f32_16x16x4_f32)
  v8f acc = {};
  int s0 = 0;
  while (s0 < S_) {
    // Next window boundary strictly greater than s0 (uniform scalar).
    int c1 = S_;
    if (t0 > s0 && t0 < c1) c1 = t0;
    if (t1 > s0 && t1 < c1) c1 = t1;
    if (l0 > s0 && l0 < c1) c1 = l0;
    if (l1 > s0 && l1 < c1) c1 = l1;
    const int cend = c1 & ~3;           // aligned end of constant-weight run

    if (cend > s0) {
      // Weight constant over [s0, cend) for this lane's A row.
      const float w = ((unsigned)(s0 - lo) < span) ? 1.0f : 0.0f;
      v2f a; a.x = w; a.y = w;
      #pragma unroll 4
      for (; s0 < cend; s0 += 4) {
        v2f bm;
        bm.x = __builtin_nontemporal_load(p);        // K = koff
        bm.y = __builtin_nontemporal_load(p + H_);   // K = koff + 1
        p += 4 * H_;
        acc = __builtin_amdgcn_wmma_f32_16x16x4_f32(false, a, false, bm,
                                                    (short)0, acc, false, false);
      }
    }
    if (s0 < c1 && s0 < S_) {
      // Boundary-straddling K-step: per-element weights (always correct).
      const int sa = s0 + koff;
      v2f a, bm;
      a.x = ((unsigned)(sa     - lo) < span) ? 1.0f : 0.0f;
      a.y = ((unsigned)(sa + 1 - lo) < span) ? 1.0f : 0.0f;
      bm.x = __builtin_nontemporal_load(p);
      bm.y = __builtin_nontemporal_load(p + H_);
      p += 4 * H_;
      acc = __builtin_amdgcn_wmma_f32_16x16x4_f32(false, a, false, bm,
                                                  (short)0, acc, false, false);
      s0 += 4;
    }
  }
  // Lanes 0-15: acc[0] = D row 0 (title sum), acc[1] = D row 1 (lead sum).
  tsum = acc[0];
  lsum = acc[1];
#else
  // Fallback: per-lane partial sums with identical memory traffic, combined
  // across lane halves (K split) with a cross-half shuffle.
  float tp = 0.0f, lp = 0.0f;
  for (int s0 = 0; s0 < S_; s0 += 4) {
    const int sa = s0 + koff;
    const float v0 = __builtin_nontemporal_load(p);
    const float v1 = __builtin_nontemporal_load(p + H_);
    p += 4 * H_;
    if ((unsigned)(sa     - t0) < (unsigned)(t1 - t0)) tp += v0;
    if ((unsigned)(sa + 1 - t0) < (unsigned)(t1 - t0)) tp += v1;
    if ((unsigned)(sa     - l0) < (unsigned)(l1 - l0)) lp += v0;
    if ((unsigned)(sa + 1 - l0) < (unsigned)(l1 - l0)) lp += v1;
  }
  tsum = tp + __shfl_xor(tp, 16, 32);
  lsum = lp + __shfl_xor(lp, 16, 32);
#endif

  if (lane < 16) {
    const float cls  = X[(size_t)b * S_ * H_ + hbase + lane];
    const float tout = (tcnt > 0) ? tsum * (1.0f / (float)tcnt) : cls;
    const float lout = (lcnt > 0) ? lsum * (1.0f / (float)lcnt) : cls;
    out[(size_t)b * H_ + hbase + lane]        = tout;   // output 0: title means
    out[(size_t)(B_ + b) * H_ + hbase + lane] = lout;   // output 1: lead means
  }
}

// ---------------------------------------------------------------------------
extern "C" void kernel_launch(void* const* d_in, const int* in_sizes, int n_in,
                              void* d_out, int out_size, void* d_ws, size_t ws_size,
                              hipStream_t stream) {
  (void)in_sizes; (void)n_in; (void)out_size; (void)ws_size;
  const float* hs  = (const float*)d_in[0];  // hidden_states (B,S,H) f32
  const int*   ids = (const int*)d_in[1];    // input_ids (B,S) int
  const int*   am  = (const int*)d_in[2];    // attention_mask (B,S) int
  const int*   sep = (const int*)d_in[3];    // sep_token_id scalar
  float*       out = (float*)d_out;          // (2,B,H) f32
  int*         meta = (int*)d_ws;            // 128 * 8 ints

  seg_meta_kernel<<<1, B_, 0, stream>>>(ids, am, sep, meta);

  dim3 grid(H_ / 16, B_);
  seg_pool_wmma_kernel<<<grid, 32, 0, stream>>>(hs, meta, out);
}